// Meta_LearnerFF_42786464203509
// MI455X (gfx1250) — compile-verified
//
#include <hip/hip_runtime.h>
#include <hip/hip_bf16.h>

typedef float v2f __attribute__((ext_vector_type(2)));
typedef float v8f __attribute__((ext_vector_type(8)));

#define Hn    512
#define WIDTH 10
#define DIN   4096
#define DTOT  (WIDTH * Hn + Hn)   // 5632
#define NROW  (Hn * WIDTH)        // 5120

// ---------------------------------------------------------------------------
// K1: f1 = relu(W1_flat(5120x4096) @ x + b1) via V_WMMA_F32_16X16X4_F32.
//
// Block = 128 threads = 4 waves; one 16-row block of W1 per block.
// Each wave owns a 1024-wide K slice (K-split x4 -> 1280 waves total), and
// keeps 4 independent accumulators so consecutive WMMAs have no D->C chain
// (4-way WMMA ILP per wave). Per-wave partial row sums (N=0 column: lane 0 =
// rows 0..7, lane 16 = rows 8..15) are reduced across the 4 waves via LDS.
//
// A(16x4) f32 layout: lanes 0-15 hold K0/K1 for M=lane, lanes 16-31 hold
// K2/K3 => per-lane b64 load at W1[row, k0 + (lane>>4)*2]. B(4x16) broadcasts
// the same x chunk into all 16 columns with the identical load pattern.
// ---------------------------------------------------------------------------
__global__ __launch_bounds__(128) void k_f1_wmma(const float* __restrict__ W1,
                                                 const float* __restrict__ x,
                                                 const float* __restrict__ b1,
                                                 float* __restrict__ f1) {
  __shared__ float partial[4][16];
  const int tid  = threadIdx.x;
  const int wave = tid >> 5;                    // 0..3 : K slice
  const int lane = tid & 31;
  const int blk  = blockIdx.x;                  // 0..319 : 16-row block
  const int row  = blk * 16 + (lane & 15);
  const int koff = (lane >> 4) * 2;             // 0 or 2
  const int kbase = wave * (DIN / 4);           // 1024 floats per wave
  const float* __restrict__ wrow = W1 + (size_t)row * DIN + kbase + koff;
  const float* __restrict__ xp   = x + kbase + koff;

  v8f c0 = {}, c1 = {}, c2 = {}, c3 = {};
#pragma unroll 4
  for (int k0 = 0; k0 < DIN / 4; k0 += 16) {
    v2f a0 = *(const v2f*)(wrow + k0);
    v2f b0 = *(const v2f*)(xp + k0);
    v2f a1 = *(const v2f*)(wrow + k0 + 4);
    v2f b1v = *(const v2f*)(xp + k0 + 4);
    v2f a2 = *(const v2f*)(wrow + k0 + 8);
    v2f b2v = *(const v2f*)(xp + k0 + 8);
    v2f a3 = *(const v2f*)(wrow + k0 + 12);
    v2f b3v = *(const v2f*)(xp + k0 + 12);
    c0 = __builtin_amdgcn_wmma_f32_16x16x4_f32(false, a0, false, b0,
                                               (short)0, c0, false, false);
    c1 = __builtin_amdgcn_wmma_f32_16x16x4_f32(false, a1, false, b1v,
                                               (short)0, c1, false, false);
    c2 = __builtin_amdgcn_wmma_f32_16x16x4_f32(false, a2, false, b2v,
                                               (short)0, c2, false, false);
    c3 = __builtin_amdgcn_wmma_f32_16x16x4_f32(false, a3, false, b3v,
                                               (short)0, c3, false, false);
  }
  c0 += c1;
  c2 += c3;
  c0 += c2;

  // N=0 column of C lives in lanes 0 (M=0..7) and 16 (M=8..15)
  if ((lane & 15) == 0) {
    const int rsub = (lane >> 4) * 8;
#pragma unroll
    for (int i = 0; i < 8; ++i) partial[wave][rsub + i] = c0[i];
  }
  __syncthreads();
  if (tid < 16) {
    const int r = blk * 16 + tid;
    float v = partial[0][tid] + partial[1][tid] + partial[2][tid] +
              partial[3][tid] + b1[r];
    f1[r] = v > 0.0f ? v : 0.0f;
  }
}

// ---------------------------------------------------------------------------
// K2: F[h,v] = relu(W2[h] (10x10) @ f1[h] + b2[h]); featvec = [F_flat, hidden]
// ---------------------------------------------------------------------------
__global__ __launch_bounds__(32) void k_f2(const float* __restrict__ W2,
                                           const float* __restrict__ b2,
                                           const float* __restrict__ hidden,
                                           const float* __restrict__ f1,
                                           float* __restrict__ feat) {
  const int h = blockIdx.x;
  const int v = threadIdx.x;
  if (v < WIDTH) {
    const float* __restrict__ w = W2 + (size_t)h * (WIDTH * WIDTH) + v * WIDTH;
    const float* __restrict__ f = f1 + h * WIDTH;
    float s = b2[h * WIDTH + v];
#pragma unroll
    for (int j = 0; j < WIDTH; ++j) s += w[j] * f[j];
    feat[h * WIDTH + v] = s > 0.0f ? s : 0.0f;
  }
  if (v == 31) feat[NROW + h] = hidden[h];
}

// ---------------------------------------------------------------------------
// K3: z[k] = sum_j Wi[k,j]*mask[k,j]*feat[j] + bi[k]; h_t, t, grad_bi,
//     grad_hidden[k] = t[k]*Wm[k,5120+k]
// ---------------------------------------------------------------------------
__global__ __launch_bounds__(256) void k_z(const float* __restrict__ Wi,
                                           const float* __restrict__ bi,
                                           const float* __restrict__ mask,
                                           const float* __restrict__ feat,
                                           float* __restrict__ ht_out,
                                           float* __restrict__ gh_out,
                                           float* __restrict__ gbi_out,
                                           float* __restrict__ t_ws) {
  __shared__ float red[256];
  const int k = blockIdx.x;
  const int tid = threadIdx.x;
  const float* __restrict__ wr = Wi + (size_t)k * DTOT;
  const float* __restrict__ mr = mask + (size_t)k * DTOT;
  float s = 0.0f;
  for (int j = tid; j < DTOT; j += 256) s += wr[j] * mr[j] * feat[j];
  red[tid] = s;
  __syncthreads();
  for (int off = 128; off > 0; off >>= 1) {
    if (tid < off) red[tid] += red[tid + off];
    __syncthreads();
  }
  if (tid == 0) {
    float z = red[0] + bi[k];
    float h = tanhf(z);
    float t = 1.0f - h * h;
    ht_out[k] = h;
    gbi_out[k] = t;
    t_ws[k] = t;
    const int j = NROW + k;
    gh_out[k] = t * wr[j] * mr[j];
  }
}

// ---------------------------------------------------------------------------
// K4: y = sum(pp * h_t)
// ---------------------------------------------------------------------------
__global__ __launch_bounds__(512) void k_y(const float* __restrict__ pp,
                                           const float* __restrict__ ht,
                                           float* __restrict__ y) {
  __shared__ float red[512];
  const int tid = threadIdx.x;
  red[tid] = pp[tid] * ht[tid];
  __syncthreads();
  for (int off = 256; off > 0; off >>= 1) {
    if (tid < off) red[tid] += red[tid + off];
    __syncthreads();
  }
  if (tid == 0) y[0] = red[0];
}

// ---------------------------------------------------------------------------
// K5: per-column backward through the tiny 10x10 stage.
//   g_F[k,v]   = t[k] * Wi[k, k*10+v] * mask[k, k*10+v]
//   g_pre2     = g_F * (F>0);  grad_b2 = g_pre2
//   grad_W2[k,v,w] = g_pre2[v] * f1[k,w]
//   g_f1[k,w]  = sum_v W2[k,v,w]*g_pre2[v]
//   g_pre1     = g_f1 * (f1>0); grad_b1 = g_pre1 (also saved to ws for K6)
// ---------------------------------------------------------------------------
__global__ __launch_bounds__(128) void k_bwd_col(const float* __restrict__ Wi,
                                                 const float* __restrict__ mask,
                                                 const float* __restrict__ W2,
                                                 const float* __restrict__ feat,
                                                 const float* __restrict__ f1,
                                                 const float* __restrict__ t_ws,
                                                 float* __restrict__ gb2,
                                                 float* __restrict__ gW2,
                                                 float* __restrict__ gb1,
                                                 float* __restrict__ g1_ws) {
  __shared__ float g2s[WIDTH];
  __shared__ float f1s[WIDTH];
  const int k = blockIdx.x;
  const int tid = threadIdx.x;
  const float t = t_ws[k];
  if (tid < WIDTH) {
    const size_t j = (size_t)k * DTOT + (size_t)k * WIDTH + tid;  // own block
    const float gF = t * Wi[j] * mask[j];
    const float Fv = feat[k * WIDTH + tid];
    const float g = (Fv > 0.0f) ? gF : 0.0f;
    g2s[tid] = g;
    gb2[k * WIDTH + tid] = g;
    f1s[tid] = f1[k * WIDTH + tid];
  }
  __syncthreads();
  if (tid < WIDTH * WIDTH) {
    const int v = tid / WIDTH, w = tid % WIDTH;
    gW2[(size_t)k * (WIDTH * WIDTH) + tid] = g2s[v] * f1s[w];
  }
  if (tid < WIDTH) {
    float s = 0.0f;
#pragma unroll
    for (int v = 0; v < WIDTH; ++v)
      s += W2[(size_t)k * (WIDTH * WIDTH) + v * WIDTH + tid] * g2s[v];
    const float g = (f1s[tid] > 0.0f) ? s : 0.0f;
    gb1[k * WIDTH + tid] = g;
    g1_ws[k * WIDTH + tid] = g;
  }
}

// ---------------------------------------------------------------------------
// K6: grad_W1[r,d] = g_pre1[r] * x[d]  (84 MB streaming write, float4)
// grid = 5120 rows * 4 sub-blocks; each block covers 1024 d (256 thr x 4)
// ---------------------------------------------------------------------------
__global__ __launch_bounds__(256) void k_gw1(const float* __restrict__ g1,
                                             const float* __restrict__ x,
                                             float* __restrict__ gW1) {
  const int bid = blockIdx.x;
  const int r = bid >> 2;
  const int d = ((bid & 3) << 10) + (threadIdx.x << 2);
  const float g = g1[r];
  const float4 xv = *(const float4*)(x + d);
  float4 o;
  o.x = g * xv.x; o.y = g * xv.y; o.z = g * xv.z; o.w = g * xv.w;
  *(float4*)(gW1 + (size_t)r * DIN + d) = o;
}

// ---------------------------------------------------------------------------
// K7: grad_Wi[k,j] = t[k] * mask[k,j] * feat[j]
// grid = 512 * 22 (5632 = 22*256)
// ---------------------------------------------------------------------------
__global__ __launch_bounds__(256) void k_gwi(const float* __restrict__ t_ws,
                                             const float* __restrict__ mask,
                                             const float* __restrict__ feat,
                                             float* __restrict__ gWi) {
  const int bid = blockIdx.x;
  const int k = bid / 22;
  const int j = (bid % 22) * 256 + threadIdx.x;
  gWi[(size_t)k * DTOT + j] = t_ws[k] * mask[(size_t)k * DTOT + j] * feat[j];
}

// ---------------------------------------------------------------------------
extern "C" void kernel_launch(void* const* d_in, const int* in_sizes, int n_in,
                              void* d_out, int out_size, void* d_ws, size_t ws_size,
                              hipStream_t stream) {
  const float* x      = (const float*)d_in[0];
  const float* hidden = (const float*)d_in[1];
  const float* pp     = (const float*)d_in[2];
  const float* W1     = (const float*)d_in[3];
  const float* b1     = (const float*)d_in[4];
  const float* W2     = (const float*)d_in[5];
  const float* b2     = (const float*)d_in[6];
  const float* Wi     = (const float*)d_in[7];
  const float* bi     = (const float*)d_in[8];
  const float* mask   = (const float*)d_in[9];

  float* out = (float*)d_out;
  // output offsets (return order: y, h_t, grad_hidden, gW1, gb1, gW2, gb2, gWi, gbi)
  float* o_y   = out;
  float* o_ht  = out + 1;
  float* o_gh  = out + 1 + Hn;
  float* o_gW1 = out + 1 + 2 * Hn;
  float* o_gb1 = o_gW1 + (size_t)NROW * DIN;
  float* o_gW2 = o_gb1 + NROW;
  float* o_gb2 = o_gW2 + (size_t)Hn * WIDTH * WIDTH;
  float* o_gWi = o_gb2 + NROW;
  float* o_gbi = o_gWi + (size_t)Hn * DTOT;

  // workspace layout (floats): f1[5120] | feat[5632] | t[512] | g1[5120]
  float* ws_f1   = (float*)d_ws;
  float* ws_feat = ws_f1 + NROW;
  float* ws_t    = ws_feat + DTOT;
  float* ws_g1   = ws_t + Hn;

  k_f1_wmma<<<NROW / 16, 128, 0, stream>>>(W1, x, b1, ws_f1);
  k_f2<<<Hn, 32, 0, stream>>>(W2, b2, hidden, ws_f1, ws_feat);
  k_z<<<Hn, 256, 0, stream>>>(Wi, bi, mask, ws_feat, o_ht, o_gh, o_gbi, ws_t);
  k_y<<<1, 512, 0, stream>>>(pp, o_ht, o_y);
  k_bwd_col<<<Hn, 128, 0, stream>>>(Wi, mask, W2, ws_feat, ws_f1, ws_t,
                                    o_gb2, o_gW2, o_gb1, ws_g1);
  k_gw1<<<NROW * 4, 256, 0, stream>>>(ws_g1, x, o_gW1);
  k_gwi<<<Hn * 22, 256, 0, stream>>>(ws_t, mask, ws_feat, o_gWi);
}